// Decompose_81793357185443
// MI455X (gfx1250) — compile-verified
//
#include <hip/hip_runtime.h>
#include <math.h>
#include <stdint.h>

#define NODES 20000
#define EDGES 200000
#define EDIM  128
#define ROW   256          // 2 channels * 128 dims, floats per node
#define TILE  256          // edges staged per block
#define BLOCK 256          // 8 wave32s
#define WAVES (BLOCK / 32)

typedef __attribute__((address_space(1))) int* gi32p;   // global AS int*
typedef __attribute__((address_space(3))) int* li32p;   // LDS AS int*

// ---- CDNA5 async global->LDS load of one dword (ASYNCcnt-tracked) ----------
__device__ __forceinline__ void async_ld_i32(const int* gsrc, int* lds_dst) {
#if __has_builtin(__builtin_amdgcn_global_load_async_to_lds_b32)
    __builtin_amdgcn_global_load_async_to_lds_b32(
        (gi32p)gsrc, (li32p)lds_dst, 0, 0);
#else
    // GV mode: VDST = LDS byte address, VADDR = 64-bit global address
    uint32_t lds_addr = (uint32_t)(uintptr_t)(void*)lds_dst;
    asm volatile("global_load_async_to_lds_b32 %0, %1, off"
                 :: "v"(lds_addr), "v"(gsrc)
                 : "memory");
#endif
}

__device__ __forceinline__ void wait_async0() {
#if __has_builtin(__builtin_amdgcn_s_wait_asynccnt)
    __builtin_amdgcn_s_wait_asynccnt(0);
#else
    asm volatile("s_wait_asynccnt 0" ::: "memory");
#endif
}

// Relaxed agent-scope f32 atomic add, result unused -> global_atomic_add_f32
// (no-return, STOREcnt-tracked, resolves in L2 where the accumulator lives).
__device__ __forceinline__ void atom_add_f32(float* p, float v) {
    __hip_atomic_fetch_add(p, v, __ATOMIC_RELAXED, __HIP_MEMORY_SCOPE_AGENT);
}

// ---------------------------------------------------------------------------
__global__ void __launch_bounds__(BLOCK)
zero_kernel(float* __restrict__ out, float* __restrict__ cnt) {
    int i = blockIdx.x * BLOCK + threadIdx.x;
    if (i < NODES * ROW) out[i] = 0.0f;
    if (i < NODES)       cnt[i] = 0.0f;
}

__global__ void __launch_bounds__(BLOCK)
scatter_kernel(const float* __restrict__ feat,
               const float* __restrict__ decomp_l,
               const float* __restrict__ decomp_r,
               const float* __restrict__ lb,
               const float* __restrict__ rb,
               const int*   __restrict__ src,
               const int*   __restrict__ dst,
               const int*   __restrict__ pos,
               float*       out,
               float*       cnt) {
    __shared__ float s_scale[2 * EDIM];   // [pos][dim] = sigmoid(decomp)
    __shared__ float s_bias [2 * EDIM];   // [pos][dim]
    __shared__ int   s_src[TILE];
    __shared__ int   s_dst[TILE];
    __shared__ int   s_pos[TILE];

    const int tid = threadIdx.x;

    // Per-block parameter staging: sigmoid once, kept in LDS.
    if (tid < EDIM) {
        s_scale[tid]        = 1.0f / (1.0f + __expf(-decomp_l[tid]));
        s_scale[EDIM + tid] = 1.0f / (1.0f + __expf(-decomp_r[tid]));
        s_bias [tid]        = lb[tid];
        s_bias [EDIM + tid] = rb[tid];
    }

    // Async-stage this block's edge-index tile into LDS (coalesced b32s).
    const int tile_base = blockIdx.x * TILE;
    {
        int e  = tile_base + tid;
        int ce = (e < EDGES) ? e : (EDGES - 1);   // keep addresses valid
        async_ld_i32(src + ce, &s_src[tid]);
        async_ld_i32(dst + ce, &s_dst[tid]);
        async_ld_i32(pos + ce, &s_pos[tid]);
    }
    wait_async0();
    __syncthreads();

    const int lane = tid & 31;
    const int wave = tid >> 5;
    const int sub  = lane * 8;            // element offset inside the 256-row
    const int dim  = sub & (EDIM - 1);    // scale/bias base (channel-invariant)

    const int n_tile = (EDGES - tile_base < TILE) ? (EDGES - tile_base) : TILE;

    for (int i = wave; i < n_tile; i += WAVES) {
        const int s = s_src[i];
        const int d = s_dst[i];
        const int p = s_pos[i];

        // Prefetch the next edge's gathered source row (global_prefetch_b8).
        const int ni = i + WAVES;
        if (ni < n_tile)
            __builtin_prefetch(feat + (size_t)s_src[ni] * ROW + sub, 0, 0);

        const float* fr = feat + (size_t)s * ROW + sub;
        float*       od = out  + (size_t)d * ROW + sub;

        const float4 a = *(const float4*)(fr);
        const float4 b = *(const float4*)(fr + 4);

        const float* sc = s_scale + p * EDIM + dim;
        const float* bi = s_bias  + p * EDIM + dim;

        atom_add_f32(od + 0, fmaf(a.x, sc[0], bi[0]));
        atom_add_f32(od + 1, fmaf(a.y, sc[1], bi[1]));
        atom_add_f32(od + 2, fmaf(a.z, sc[2], bi[2]));
        atom_add_f32(od + 3, fmaf(a.w, sc[3], bi[3]));
        atom_add_f32(od + 4, fmaf(b.x, sc[4], bi[4]));
        atom_add_f32(od + 5, fmaf(b.y, sc[5], bi[5]));
        atom_add_f32(od + 6, fmaf(b.z, sc[6], bi[6]));
        atom_add_f32(od + 7, fmaf(b.w, sc[7], bi[7]));

        if (lane == 0) atom_add_f32(cnt + d, 1.0f);
    }
}

__global__ void __launch_bounds__(BLOCK)
norm_kernel(float* __restrict__ out, const float* __restrict__ cnt) {
    int i = blockIdx.x * BLOCK + threadIdx.x;
    if (i < NODES * ROW) {
        float c = cnt[i >> 8];                // i / ROW
        out[i] *= 1.0f / fmaxf(c, 1.0f);
    }
}

// ---------------------------------------------------------------------------
extern "C" void kernel_launch(void* const* d_in, const int* in_sizes, int n_in,
                              void* d_out, int out_size, void* d_ws, size_t ws_size,
                              hipStream_t stream) {
    const float* feat     = (const float*)d_in[0];
    const float* decomp_l = (const float*)d_in[1];
    const float* decomp_r = (const float*)d_in[2];
    const float* lb       = (const float*)d_in[3];
    const float* rb       = (const float*)d_in[4];
    const int*   src      = (const int*)d_in[5];
    const int*   dst      = (const int*)d_in[6];
    const int*   pos      = (const int*)d_in[7];

    float* out = (float*)d_out;            // [NODES, 2, 128] f32
    float* cnt = (float*)d_ws;             // NODES floats of scratch

    const int elem_blocks = (NODES * ROW + BLOCK - 1) / BLOCK;
    zero_kernel<<<elem_blocks, BLOCK, 0, stream>>>(out, cnt);

    const int tiles = (EDGES + TILE - 1) / TILE;
    scatter_kernel<<<tiles, BLOCK, 0, stream>>>(feat, decomp_l, decomp_r,
                                                lb, rb, src, dst, pos,
                                                out, cnt);

    norm_kernel<<<elem_blocks, BLOCK, 0, stream>>>(out, cnt);
}